// ODEFunction_14946486190216
// MI455X (gfx1250) — compile-verified
//
#include <hip/hip_runtime.h>

#define D_FEAT 64

__global__ __launch_bounds__(256) void zero_f4_kernel(float4* __restrict__ out, int n4) {
    int i = blockIdx.x * blockDim.x + threadIdx.x;
    if (i < n4) out[i] = float4{0.f, 0.f, 0.f, 0.f};
}

// One wave (32 lanes) processes 32 edges.
// Phase 0: lane e loads edge metadata + prefetches its gather row (256B = 2 lines).
// Phase 1: issue ALL 32 float2 gathers back-to-back (32 in-flight loads/wave,
//          SGPR-base addressing from v_readlane'd col), single wait.
// Phase 2: issue all 64 no-return global_atomic_add_f32 back-to-back.
__global__ __launch_bounds__(256) void coo_spmm_scatter_kernel(
    const float* __restrict__ x,
    const int*   __restrict__ rows,
    const int*   __restrict__ cols,
    const float* __restrict__ vals,
    float*       __restrict__ out,
    int E) {
    const int lane = threadIdx.x & 31;
    const int wave = (blockIdx.x * blockDim.x + threadIdx.x) >> 5;
    const int base = wave * 32;
    if (base >= E) return;

    const int  e     = base + lane;
    const bool valid = (e < E);

    const int   r = valid ? rows[e] : 0;
    const int   c = valid ? cols[e] : 0;
    const float w = valid ? vals[e] : 0.0f;

    // Pre-touch all 32 gather rows so the random L2 reads start early.
    if (valid) {
        const float* p = x + (size_t)c * D_FEAT;
        __builtin_prefetch(p,      0, 3);
        __builtin_prefetch(p + 32, 0, 3);
    }

    const int fo = lane * 2;  // each lane owns 2 of the 64 features

    if (base + 32 <= E) {
        // ---- Phase 1: batch all gathers (maximize MLP, minimize waits) ----
        float2 v[32];
#pragma unroll
        for (int j = 0; j < 32; ++j) {
            const int cj = __builtin_amdgcn_readlane(c, j);
            v[j] = *reinterpret_cast<const float2*>(x + (size_t)cj * D_FEAT + fo);
        }
        // ---- Phase 2: batch all scatter atomics ----
#pragma unroll
        for (int j = 0; j < 32; ++j) {
            const int   rj = __builtin_amdgcn_readlane(r, j);
            const float wj = __uint_as_float(
                __builtin_amdgcn_readlane(__float_as_uint(w), j));
            float* o = out + (size_t)rj * D_FEAT + fo;
            unsafeAtomicAdd(o + 0, wj * v[j].x);
            unsafeAtomicAdd(o + 1, wj * v[j].y);
        }
    } else {
        const int n = E - base;
        for (int j = 0; j < n; ++j) {
            const int   rj = __builtin_amdgcn_readlane(r, j);
            const int   cj = __builtin_amdgcn_readlane(c, j);
            const float wj = __uint_as_float(
                __builtin_amdgcn_readlane(__float_as_uint(w), j));
            const float2 v =
                *reinterpret_cast<const float2*>(x + (size_t)cj * D_FEAT + fo);
            float* o = out + (size_t)rj * D_FEAT + fo;
            unsafeAtomicAdd(o + 0, wj * v.x);
            unsafeAtomicAdd(o + 1, wj * v.y);
        }
    }
}

extern "C" void kernel_launch(void* const* d_in, const int* in_sizes, int n_in,
                              void* d_out, int out_size, void* d_ws, size_t ws_size,
                              hipStream_t stream) {
    // setup_inputs order: t(scalar), x[f32], rows[i32], cols[i32], edge_vals[f32]
    const float* x    = (const float*)d_in[1];
    const int*   rows = (const int*)  d_in[2];
    const int*   cols = (const int*)  d_in[3];
    const float* vals = (const float*)d_in[4];
    float*       out  = (float*)d_out;

    const int E = in_sizes[2];

    // 1) Zero the (poisoned) output buffer.
    const int n4          = out_size / 4;
    const int zero_blocks = (n4 + 255) / 256;
    zero_f4_kernel<<<zero_blocks, 256, 0, stream>>>((float4*)out, n4);

    // 2) Scatter-add: one wave per 32 edges, 8 waves per 256-thread block.
    const int waves  = (E + 31) / 32;
    const int blocks = (waves + 7) / 8;
    coo_spmm_scatter_kernel<<<blocks, 256, 0, stream>>>(x, rows, cols, vals, out, E);
}